// FeatureClustering_37580963840133
// MI455X (gfx1250) — compile-verified
//
#include <hip/hip_runtime.h>
#include <math.h>

// FeatureClustering on MI455X (gfx1250, wave32).
// R=2M rows x E=64 feats fp32 (512 MB stream), K=8 clusters, B=64K segments.
// Memory-bound: ~22us floor at 23.3 TB/s. The [R,64]x[64,8] projection GEMM
// runs on V_WMMA_F32_16X16X4_F32 (16 WMMAs per 16-row tile), VALU does the
// EMG (erfc/log/exp) math, segment reduction via global f32 atomics into L2.

typedef float v2f __attribute__((ext_vector_type(2)));
typedef float v8f __attribute__((ext_vector_type(8)));

#define LOG_TWO_F 0.6931471805599453f

__global__ void fc_zero_ws(float* __restrict__ ws, int n) {
  int i = blockIdx.x * blockDim.x + threadIdx.x;
  if (i < n) ws[i] = 0.0f;
}

__global__ __launch_bounds__(256)
void fc_main(const float* __restrict__ alt,        // [R,64]
             const float* __restrict__ centroid,   // [64]
             const float* __restrict__ na_ls_p,    // [1]
             const float* __restrict__ dirs,       // [8,64]
             const float* __restrict__ mu_k,       // [8]
             const float* __restrict__ log_sigma_k,// [8]
             const float* __restrict__ log_lambda_k,//[8]
             const float* __restrict__ art_ls_k,   // [8]
             const int*   __restrict__ seg,        // [R] sorted
             float* __restrict__ ws_na,            // [B]
             float* __restrict__ ws_art,           // [B,8]
             int R)
{
  const int lane = threadIdx.x & 31;
  const int h    = lane >> 4;   // lane half (selects K-pair / row half)
  const int n    = lane & 15;   // row-in-tile for A loads; cluster col for D
  const int waveId = (blockIdx.x * blockDim.x + threadIdx.x) >> 5;
  const int nWaves = (gridDim.x * blockDim.x) >> 5;

  const float na_ls   = na_ls_p[0];
  const float inv2nav = 0.5f * __expf(-2.0f * na_ls); // 1/(2*exp(2*ls))
  const float na_c    = -64.0f * na_ls;

  // Per-lane centroid slices for the 16 K-chunks (col = 4c + 2h, 2 floats).
  v2f cen[16];
  #pragma unroll
  for (int c = 0; c < 16; ++c) {
    int col = 4 * c + 2 * h;
    cen[c].x = centroid[col];
    cen[c].y = centroid[col + 1];
  }

  // B-matrix slices: B_chunk[kk][col] = unit[col][4c+kk], cols 8..15 zero.
  // Layout mirrors A: VGPR v, lane L -> Bmat[kk = v + 2*(L>>4)][N = L&15].
  float invn = 0.0f;
  if (n < 8) {
    float s = 0.0f;
    #pragma unroll 8
    for (int e = 0; e < 64; ++e) { float v = dirs[n * 64 + e]; s += v * v; }
    invn = rsqrtf(s);
  }
  v2f bmat[16];
  #pragma unroll
  for (int c = 0; c < 16; ++c) {
    int e = 4 * c + 2 * h;
    float bx = 0.0f, by = 0.0f;
    if (n < 8) {
      bx = dirs[n * 64 + e] * invn;
      by = dirs[n * 64 + e + 1] * invn;
    }
    bmat[c].x = bx; bmat[c].y = by;
  }

  // Per-lane EMG / orthogonal constants for cluster n (n<8).
  float c0 = 0.f, c1 = 0.f, A1 = 0.f, inv_s = 0.f, muv = 0.f, o0 = 0.f, oinv = 0.f;
  if (n < 8) {
    muv = mu_k[n];
    float ls  = log_sigma_k[n];
    float ll  = log_lambda_k[n];
    float var = __expf(2.0f * ls);
    float lam = __expf(ll);
    A1    = muv + lam * var;          // mu + lambda*var
    inv_s = rsqrtf(2.0f * var);       // 1/sqrt(2*var)
    c0    = ll - LOG_TWO_F;
    c1    = 0.5f * lam;
    float als = art_ls_k[n];
    o0   = -63.0f * als;              // -(E-1)*art_log_stdev
    oinv = 0.5f * __expf(-2.0f * als);
  }

  const int nTiles = R >> 4;
  for (int tile = waveId; tile < nTiles; tile += nWaves) {
    const int r0 = tile << 4;
    const float* rowp = alt + (size_t)(r0 + n) * 64;

    v8f acc = {0.f, 0.f, 0.f, 0.f, 0.f, 0.f, 0.f, 0.f};
    float sq = 0.0f;
    #pragma unroll
    for (int c = 0; c < 16; ++c) {
      // A-slab: lane (h,n) holds delta[row=n][4c+2h .. 4c+2h+1]
      v2f x = *(const v2f*)(rowp + 4 * c + 2 * h);
      v2f d; d.x = x.x - cen[c].x; d.y = x.y - cen[c].y;
      sq = fmaf(d.x, d.x, sq);
      sq = fmaf(d.y, d.y, sq);
      acc = __builtin_amdgcn_wmma_f32_16x16x4_f32(
          false, d, false, bmat[c], (short)0, acc, false, false);
    }
    // Full |delta|^2 per row: two lane-halves hold complementary columns.
    float sqf = sq + __shfl_xor(sq, 16, 32);
    int segRow = seg[r0 + n]; // all lanes: segment of row n (dup across halves)

    if (lane < 16)
      atomicAdd(&ws_na[segRow], na_c - sqf * inv2nav);

    // D: VGPR j, lane L -> dot[row = j + 8*(L>>4)][cluster = L&15]
    #pragma unroll
    for (int j = 0; j < 8; ++j) {
      const int src = j + 8 * h;               // lane holding row j+8h data
      float dot = acc[j];
      float sqr = __shfl(sqf, src, 32);
      int   sg  = __shfl(segRow, src, 32);
      if (n < 8) {
        float orth = fmaxf(sqr - dot * dot, 0.0f);
        float oll  = o0 - orth * oinv;
        float pll  = c0 + __logf(erfcf((A1 - dot) * inv_s))
                   + c1 * (A1 + muv - 2.0f * dot);
        atomicAdd(&ws_art[(size_t)sg * 8 + n], oll + pll);
      }
    }
  }
}

__global__ void fc_final(const float* __restrict__ ws_na,
                         const float* __restrict__ ws_art,
                         const float* __restrict__ wpre,
                         float* __restrict__ out, int B)
{
  int b = blockIdx.x * blockDim.x + threadIdx.x;
  if (b >= B) return;
  float w[8], wmax = -INFINITY;
  #pragma unroll
  for (int k = 0; k < 8; ++k) { w[k] = wpre[k]; wmax = fmaxf(wmax, w[k]); }
  float sw = 0.0f;
  #pragma unroll
  for (int k = 0; k < 8; ++k) sw += __expf(w[k] - wmax);
  float lsz = wmax + __logf(sw); // logsumexp of pre-softmax weights

  float v[8], vmax = -INFINITY;
  #pragma unroll
  for (int k = 0; k < 8; ++k) {
    v[k] = ws_art[(size_t)b * 8 + k] + (w[k] - lsz);
    vmax = fmaxf(vmax, v[k]);
  }
  float s = 0.0f;
  #pragma unroll
  for (int k = 0; k < 8; ++k) s += __expf(v[k] - vmax);
  out[b] = vmax + __logf(s) - ws_na[b];
}

extern "C" void kernel_launch(void* const* d_in, const int* in_sizes, int n_in,
                              void* d_out, int out_size, void* d_ws, size_t ws_size,
                              hipStream_t stream) {
  const float* alt      = (const float*)d_in[0];
  const float* centroid = (const float*)d_in[1];
  const float* na_ls    = (const float*)d_in[2];
  const float* dirs     = (const float*)d_in[3];
  const float* mu       = (const float*)d_in[4];
  const float* lsg      = (const float*)d_in[5];
  const float* llm      = (const float*)d_in[6];
  const float* als      = (const float*)d_in[7];
  const float* wpre     = (const float*)d_in[8];
  const int*   seg      = (const int*)d_in[9];

  const int R = in_sizes[0] / 64;
  const int B = out_size;

  float* ws_na  = (float*)d_ws;
  float* ws_art = ws_na + B;

  const int nz = B * 9;
  fc_zero_ws<<<(nz + 255) / 256, 256, 0, stream>>>(ws_na, nz);
  fc_main<<<2048, 256, 0, stream>>>(alt, centroid, na_ls, dirs, mu, lsg, llm,
                                    als, seg, ws_na, ws_art, R);
  fc_final<<<(B + 255) / 256, 256, 0, stream>>>(ws_na, ws_art, wpre,
                                                (float*)d_out, B);
}